// ChamferDistance_27307402068672
// MI455X (gfx1250) — compile-verified
//
#include <hip/hip_runtime.h>
#include <hip/hip_bf16.h>

typedef float v2f __attribute__((ext_vector_type(2)));
typedef float v8f __attribute__((ext_vector_type(8)));

#define NPTS 4096
#define NTILES (NPTS / 16)
#define NBLK_X (NTILES / 2)          // two A-tiles per wave
#define NPARTIAL (NBLK_X * 8 * 2)    // 2048 partial sums

// One wave (32 threads) per block. Each wave owns TWO 16-row tiles of the "A"
// point set and loops over all 256 16-column tiles of the "B" set. Per B tile
// a single fragment feeds two v_wmma_f32_16x16x4_f32:
//   A row  = [a0, a1, a2, 1]
//   B col  = [-2b0, -2b1, -2b2, ||b||^2]   (K=3 slot carries the y-norm)
//   C      = ||a_row||^2   -> depends only on the row => LOOP-INVARIANT,
//                             built once via shuffles, no per-tile movs.
//   D      = ||a||^2 - 2 a.b + ||b||^2
__global__ __launch_bounds__(32)
void chamfer_min_kernel(const float* __restrict__ x,
                        const float* __restrict__ y,
                        float* __restrict__ partial) {
    const int lane = threadIdx.x;      // 0..31
    const int h    = lane >> 4;        // half-wave: 0 or 1
    const int j    = lane & 15;        // row/col index within tile
    const int b    = blockIdx.y;       // batch (0..7)
    const int dir  = blockIdx.z;       // 0: A=x,B=y ; 1: A=y,B=x

    const float* __restrict__ Ap = dir ? y : x;
    const float* __restrict__ Bp = dir ? x : y;
    const size_t base = (size_t)b * 3 * NPTS;   // layout: [B, dim, N]

    // ---- Two fixed A-matrix fragments (tiles 2*bx and 2*bx+1) ----
    // A-matrix 16x4 f32: lanes 0-15 -> (K=0,K=1), lanes 16-31 -> (K=2,K=3)
    const int ar0 = (blockIdx.x * 2 + 0) * 16 + j;
    const int ar1 = (blockIdx.x * 2 + 1) * 16 + j;
    const float a00 = Ap[base + 0 * NPTS + ar0];
    const float a01 = Ap[base + 1 * NPTS + ar0];
    const float a02 = Ap[base + 2 * NPTS + ar0];
    const float a10 = Ap[base + 0 * NPTS + ar1];
    const float a11 = Ap[base + 1 * NPTS + ar1];
    const float a12 = Ap[base + 2 * NPTS + ar1];
    const float aa0 = a00 * a00 + a01 * a01 + a02 * a02;
    const float aa1 = a10 * a10 + a11 * a11 + a12 * a12;

    v2f am0, am1;
    am0.x = h ? a02 : a00;  am0.y = h ? 1.0f : a01;   // K=2:K=0 , K=3:K=1
    am1.x = h ? a12 : a10;  am1.y = h ? 1.0f : a11;

    // ---- Loop-invariant C fragments: c[v] = ||a_row||^2, row = v + 8*h.
    // Row r's norm lives in lane r (and r+16); gather once with shuffles.
    v8f c0, c1;
#pragma unroll
    for (int v = 0; v < 8; ++v) {
        const int src = 8 * h + v;
        c0[v] = __shfl(aa0, src, 32);
        c1[v] = __shfl(aa1, src, 32);
    }

    // Running per-element mins.
    v8f m0, m1;
#pragma unroll
    for (int v = 0; v < 8; ++v) { m0[v] = __builtin_inff(); m1[v] = __builtin_inff(); }

    for (int t = 0; t < NTILES; ++t) {
        const int bc = t * 16 + j;
        const float b0 = Bp[base + 0 * NPTS + bc];
        const float b1 = Bp[base + 1 * NPTS + bc];
        const float b2 = Bp[base + 2 * NPTS + bc];
        const float bb = b0 * b0 + b1 * b1 + b2 * b2;

        // B-matrix 4x16 f32: lanes 0-15 -> (K=0,K=1), lanes 16-31 -> (K=2,K=3)
        v2f bm;
        bm.x = h ? (-2.0f * b2) : (-2.0f * b0);
        bm.y = h ? bb           : (-2.0f * b1);

        v8f d0 = __builtin_amdgcn_wmma_f32_16x16x4_f32(
            false, am0, false, bm, (short)0, c0, false, false);
        v8f d1 = __builtin_amdgcn_wmma_f32_16x16x4_f32(
            false, am1, false, bm, (short)0, c1, false, false);

#pragma unroll
        for (int v = 0; v < 8; ++v) {
            m0[v] = fminf(m0[v], d0[v]);
            m1[v] = fminf(m1[v], d1[v]);
        }
    }

    // ---- Per-row min across the 16 columns held by each half-wave ----
    // D layout: element (M = v + 8*h, N = lane%16); xor masks 1,2,4,8 stay
    // inside each half and cover all N.
#pragma unroll
    for (int v = 0; v < 8; ++v) {
        float u0 = m0[v], u1 = m1[v];
        u0 = fminf(u0, __shfl_xor(u0, 1, 32));
        u1 = fminf(u1, __shfl_xor(u1, 1, 32));
        u0 = fminf(u0, __shfl_xor(u0, 2, 32));
        u1 = fminf(u1, __shfl_xor(u1, 2, 32));
        u0 = fminf(u0, __shfl_xor(u0, 4, 32));
        u1 = fminf(u1, __shfl_xor(u1, 4, 32));
        u0 = fminf(u0, __shfl_xor(u0, 8, 32));
        u1 = fminf(u1, __shfl_xor(u1, 8, 32));
        m0[v] = u0; m1[v] = u1;
    }
    // Sum of per-row mins for this half's 8 rows, both tiles...
    float s = (m0[0] + m0[1] + m0[2] + m0[3] + m0[4] + m0[5] + m0[6] + m0[7])
            + (m1[0] + m1[1] + m1[2] + m1[3] + m1[4] + m1[5] + m1[6] + m1[7]);
    // ...combine the two halves (rows 0-7 + rows 8-15 of each tile).
    s += __shfl_xor(s, 16, 32);

    if (lane == 0) {
        partial[((size_t)dir * 8 + b) * NBLK_X + blockIdx.x] = s;
    }
}

// Deterministic final reduction of the 2048 partial sums.
__global__ __launch_bounds__(256)
void chamfer_finalize_kernel(const float* __restrict__ partial,
                             float* __restrict__ out) {
    __shared__ float sm[256];
    const int t = threadIdx.x;
    float s = 0.0f;
#pragma unroll
    for (int i = 0; i < NPARTIAL / 256; ++i) s += partial[t + i * 256];
    sm[t] = s;
    __syncthreads();
    for (int stride = 128; stride > 0; stride >>= 1) {
        if (t < stride) sm[t] += sm[t + stride];
        __syncthreads();
    }
    if (t == 0) {
        // distance_b = (sum_min_x + sum_min_y)/4096 ; mean over 8 batches.
        out[0] = sm[0] * (1.0f / (4096.0f * 8.0f));
    }
}

extern "C" void kernel_launch(void* const* d_in, const int* in_sizes, int n_in,
                              void* d_out, int out_size, void* d_ws, size_t ws_size,
                              hipStream_t stream) {
    const float* x = (const float*)d_in[0];   // [8, 3, 4096]
    const float* y = (const float*)d_in[1];   // [8, 3, 4096]
    float* partial = (float*)d_ws;            // 2048 floats = 8 KB scratch
    float* out     = (float*)d_out;

    dim3 grid(NBLK_X, 8, 2);   // (A-tile pair, batch, direction)
    dim3 block(32);            // one wave32 per block (WMMA needs EXEC all-1s)
    chamfer_min_kernel<<<grid, block, 0, stream>>>(x, y, partial);
    chamfer_finalize_kernel<<<1, 256, 0, stream>>>(partial, out);
}